// GCN_Block_13056700579874
// MI455X (gfx1250) — compile-verified
//
#include <hip/hip_runtime.h>

typedef __attribute__((ext_vector_type(2))) float v2f;
typedef __attribute__((ext_vector_type(8))) float v8f;

#define N_FEAT 128
#define K_DIM  256

// ---------------- scratch zeroing (grid-stride) ----------------
__global__ void zero_f32(float* __restrict__ p, long long n) {
  long long i = (long long)blockIdx.x * blockDim.x + threadIdx.x;
  const long long stride = (long long)gridDim.x * blockDim.x;
  for (; i < n; i += stride) p[i] = 0.0f;
}

// ---------------- deg[dst] += 1 ----------------
__global__ void degree_kernel(const int* __restrict__ dst, int E,
                              float* __restrict__ deg) {
  int e = blockIdx.x * blockDim.x + threadIdx.x;
  if (e < E) atomicAdd(&deg[dst[e]], 1.0f);
}

// ---------------- dis = deg>0 ? rsqrt(deg) : 0 (in place) ----------------
__global__ void dis_kernel(float* __restrict__ deg, int n) {
  int i = blockIdx.x * blockDim.x + threadIdx.x;
  if (i < n) {
    float d = deg[i];
    deg[i] = (d > 0.0f) ? rsqrtf(d) : 0.0f;
  }
}

// ---------------- agg[dst] += norm * x[src] ; one wave32 per edge ----------------
__global__ void scatter_kernel(const float* __restrict__ x,
                               const int* __restrict__ src,
                               const int* __restrict__ dst,
                               const float* __restrict__ dis,
                               float* __restrict__ agg, int E) {
  int gid  = blockIdx.x * blockDim.x + threadIdx.x;
  int edge = gid >> 5;
  int lane = gid & 31;
  if (edge >= E) return;
  int s = src[edge];
  int d = dst[edge];
  float nrm = dis[s] * dis[d];
  // coalesced 512B gather of row x[s]; lane covers features [4*lane, 4*lane+3]
  const float4 xv = reinterpret_cast<const float4*>(x + (long long)s * N_FEAT)[lane];
  float* ap = agg + (long long)d * N_FEAT + lane * 4;
  atomicAdd(ap + 0, xv.x * nrm);
  atomicAdd(ap + 1, xv.y * nrm);
  atomicAdd(ap + 2, xv.z * nrm);
  atomicAdd(ap + 3, xv.w * nrm);
}

// ---------------- out = [x | agg] @ W via V_WMMA_F32_16X16X4_F32 ----------------
// One wave per 16-row M-tile; 8 accumulators span all N=128 columns.
// A (16x4 f32): lane = M (lane%16); VGPR0 = K {0 | 2}, VGPR1 = K {1 | 3} by lane-half.
// B (4x16 f32): lane%16 = N; VGPR0 = K {0 | 2}, VGPR1 = K {1 | 3} by lane-half.
// C/D (16x16 f32): row = vgpr + 8*(lane/16), col = lane%16.
__global__ void tag_gemm_wmma(const float* __restrict__ x,
                              const float* __restrict__ agg,
                              const float* __restrict__ W,
                              float* __restrict__ out, int n_nodes) {
  const int lane   = threadIdx.x & 31;
  const int tile   = (blockIdx.x * blockDim.x + threadIdx.x) >> 5;
  const long long m0 = (long long)tile * 16;
  if (m0 >= n_nodes) return;          // wave-uniform: surviving waves keep EXEC all-1s
  const int laneLo = lane & 15;
  const int laneHi = lane >> 4;       // 0 or 1

  v8f acc[8];
  const v8f vzero = {0.f, 0.f, 0.f, 0.f, 0.f, 0.f, 0.f, 0.f};
#pragma unroll
  for (int t = 0; t < 8; ++t) acc[t] = vzero;

  const float* __restrict__ arow_x = x   + (m0 + laneLo) * N_FEAT;
  const float* __restrict__ arow_g = agg + (m0 + laneLo) * N_FEAT;

#pragma unroll 4
  for (int kb = 0; kb < K_DIM / 4; ++kb) {
    const int k  = kb * 4;
    const int kk = k & (N_FEAT - 1);
    const float* __restrict__ hrow = (k < N_FEAT) ? arow_x : arow_g;
    // A fragment: two consecutive K values for this lane-half
    v2f a = *reinterpret_cast<const v2f*>(hrow + kk + 2 * laneHi);
#pragma unroll
    for (int t = 0; t < 8; ++t) {
      const int n = t * 16 + laneLo;
      v2f b;
      b.x = W[(k +     2 * laneHi) * N_FEAT + n];
      b.y = W[(k + 1 + 2 * laneHi) * N_FEAT + n];
      acc[t] = __builtin_amdgcn_wmma_f32_16x16x4_f32(
          /*neg_a=*/false, a, /*neg_b=*/false, b,
          /*c_mod=*/(short)0, acc[t], /*reuse_a=*/false, /*reuse_b=*/false);
    }
  }

#pragma unroll
  for (int t = 0; t < 8; ++t) {
#pragma unroll
    for (int r = 0; r < 8; ++r) {
      const long long m = m0 + r + 8 * laneHi;
      out[m * N_FEAT + t * 16 + laneLo] = acc[t][r];
    }
  }
}

extern "C" void kernel_launch(void* const* d_in, const int* in_sizes, int n_in,
                              void* d_out, int out_size, void* d_ws, size_t ws_size,
                              hipStream_t stream) {
  (void)n_in; (void)out_size; (void)ws_size;

  const float* x    = (const float*)d_in[0];   // [N,128] f32
  const int*   edge = (const int*)d_in[1];     // [2,E] (harness int mapping)
  const float* W    = (const float*)d_in[2];   // [256,128] f32
  float*       out  = (float*)d_out;           // [N,128] f32

  const int N = in_sizes[0] / N_FEAT;
  const int E = in_sizes[1] / 2;
  const int* src = edge;
  const int* dst = edge + E;

  // workspace layout: [deg/dis : N f32][pad to 256B][agg : N*128 f32]
  float* deg = (float*)d_ws;
  const size_t agg_off = ((size_t)N * sizeof(float) + 255) & ~(size_t)255;
  float* agg = (float*)((char*)d_ws + agg_off);

  const long long total_zero = (long long)(agg_off / sizeof(float)) +
                               (long long)N * N_FEAT;
  zero_f32<<<2048, 256, 0, stream>>>((float*)d_ws, total_zero);

  degree_kernel<<<(E + 255) / 256, 256, 0, stream>>>(dst, E, deg);
  dis_kernel<<<(N + 255) / 256, 256, 0, stream>>>(deg, N);

  const long long sthreads = (long long)E * 32;  // one wave per edge
  scatter_kernel<<<(unsigned)((sthreads + 255) / 256), 256, 0, stream>>>(
      x, src, dst, deg, agg, E);

  const int mtiles = (N + 15) / 16;              // 6250 waves
  const int blocks = (mtiles * 32 + 255) / 256;
  tag_gemm_wmma<<<blocks, 256, 0, stream>>>(x, agg, W, out, N);
}